// FourierPositionEmbeddingNEW_26877905338722
// MI455X (gfx1250) — compile-verified
//
#include <hip/hip_runtime.h>

typedef __attribute__((ext_vector_type(2))) float v2f;
typedef __attribute__((ext_vector_type(8))) float v8f;

#define OCH 512   // hidden channels
#define SZ  256   // spatial size (square)
#define NB  4     // batch
#define H_PER_WAVE 8

// out[b,o,h,w] = x[b,o,h,w] + sin(W[o,0]*xw(w) + W[o,1]*yh(h) + bias[o])
// One wave computes a 16(o) x 16(w) tile of the fourier term via
// V_WMMA_F32_16X16X4_F32 and streams the 4 batch planes through it.
__global__ __launch_bounds__(256) void fourier_embed_wmma(
    const float* __restrict__ x,
    const float* __restrict__ W,
    const float* __restrict__ bias,
    float* __restrict__ out)
{
    const int lane        = threadIdx.x & 31;
    const int waveInBlock = threadIdx.x >> 5;
    const int wave        = blockIdx.x * 8 + waveInBlock;

    // tile decomposition: 16 w-tiles * 32 h-blocks * 32 o-tiles = 16384 waves
    const int wt = wave & 15;          // w tile
    const int hb = (wave >> 4) & 31;   // h block (8 rows)
    const int ot = wave >> 9;          // o tile

    const int w0 = wt * 16;
    const int h0 = hb * H_PER_WAVE;
    const int o0 = ot * 16;

    const int  m      = lane & 15;
    const bool hiHalf = lane >= 16;

    // ---- A matrix 16x4 (M=o, K=4): row = [W0[o], W1[o], b[o], 0]
    // Layout (ISA 7.12.2, 32-bit A 16x4): lanes 0-15 hold K=0(K=1 in v.y),
    // lanes 16-31 hold K=2 (K=3 in v.y); M = lane%16.
    const float w0v = W[(o0 + m) * 2 + 0];
    const float w1v = W[(o0 + m) * 2 + 1];
    const float bv  = bias[o0 + m];
    v2f a;
    a.x = hiHalf ? bv   : w0v;   // K=2 : K=0
    a.y = hiHalf ? 0.0f : w1v;   // K=3 : K=1

    const float step = 2.0f / (float)(SZ - 1);
    const float xw   = (float)(w0 + m) * step - 1.0f;  // B row K=0, N = lane%16

    for (int hh = 0; hh < H_PER_WAVE; ++hh) {
        const int   h  = h0 + hh;
        const float yh = (float)h * step - 1.0f;

        // ---- B matrix 4x16 (K x N=w): col = [xw, yh, 1, 0]
        // VGPR0: lanes0-15 = K=0 row, lanes16-31 = K=2 row
        // VGPR1: lanes0-15 = K=1 row, lanes16-31 = K=3 row
        v2f bm;
        bm.x = hiHalf ? 1.0f : xw;
        bm.y = hiHalf ? 0.0f : yh;

        v8f c = {};
        // D[o][w] = W0[o]*xw + W1[o]*yh + bias[o]
        v8f d = __builtin_amdgcn_wmma_f32_16x16x4_f32(
            /*neg_a=*/false, a, /*neg_b=*/false, bm,
            /*c_mod=*/(short)0, c, /*reuse_a=*/false, /*reuse_b=*/false);

        // sin of the 16x16 tile (8 values per lane); v_sin_f32 path
        float s[8];
#pragma unroll
        for (int r = 0; r < 8; ++r) s[r] = __sinf(d[r]);

        // D layout: VGPR r, lanes 0-15 -> M=r, lanes 16-31 -> M=r+8; N=lane%16
#pragma unroll
        for (int r = 0; r < 8; ++r) {
            const int    o    = o0 + r + (hiHalf ? 8 : 0);
            const size_t base = (((size_t)o * SZ) + (size_t)h) * SZ
                              + (size_t)(w0 + m);
#pragma unroll
            for (int bi = 0; bi < NB; ++bi) {
                const size_t idx = (size_t)bi * (size_t)OCH * SZ * SZ + base;
                // streaming data (>> L2): non-temporal load + store
                const float xv = __builtin_nontemporal_load(&x[idx]);
                __builtin_nontemporal_store(xv + s[r], &out[idx]);
            }
        }
    }
}

extern "C" void kernel_launch(void* const* d_in, const int* in_sizes, int n_in,
                              void* d_out, int out_size, void* d_ws, size_t ws_size,
                              hipStream_t stream) {
    (void)in_sizes; (void)n_in; (void)out_size; (void)d_ws; (void)ws_size;
    const float* x    = (const float*)d_in[0];
    const float* W    = (const float*)d_in[1];
    const float* bias = (const float*)d_in[2];
    float*       out  = (float*)d_out;

    // 32 o-tiles * 16 w-tiles * 32 h-blocks = 16384 waves, 8 waves/block
    dim3 grid(2048), block(256);
    fourier_embed_wmma<<<grid, block, 0, stream>>>(x, W, bias, out);
}